// Model_hmm_28561532518940
// MI455X (gfx1250) — compile-verified
//
#include <hip/hip_runtime.h>

// Problem dims (fixed by the reference)
#define B_SZ  256
#define T_LEN 256
#define NS    128   // N_STATES
#define NO    64    // N_OUT

typedef __attribute__((ext_vector_type(2))) float v2f;
typedef __attribute__((ext_vector_type(8))) float v8f;

union v8fu { v8f v; float f[8]; };

// ---------------------------------------------------------------------------
// Generic WMMA f32 GEMM:  C = alpha*op(A)*op(B) + beta*E + gamma*I
//   TA/TB are compile-time transpose flags -> branch-free inner loop,
//   contiguous k-pairs load as b64, j-tiles fully unrolled (4 WMMA / k-step,
//   A-frag reused across the 4 B-frags).
//   block = 128 threads (4 waves), each block computes a 64x64 tile.
//   Batched via blockIdx.y with per-operand batch strides.
//   M,N multiples of 64; K multiple of 8 (all call sites: 64/128/256).
//   tc==1 writes C transposed (C[n*ldc+m]) to hit [B,T,*] output layouts.
// ---------------------------------------------------------------------------
template <int TA, int TB>
__global__ __launch_bounds__(128) void gemm_wmma(
    const float* __restrict__ A, int lda, long sA,
    const float* __restrict__ B, int ldb, long sB,
    const float* __restrict__ E, int lde, long sE,
    float* __restrict__ C, int ldc, long sC, int tc,
    int N, int K, float alpha, float beta, float gam)
{
  const int tn  = N >> 6;
  const int bm  = blockIdx.x / tn;
  const int bn  = blockIdx.x - bm * tn;
  const long bat = blockIdx.y;
  A += bat * sA;
  B += bat * sB;
  C += bat * sC;
  if (E) E += bat * sE;

  const int tid  = threadIdx.x;
  const int wave = tid >> 5;
  const int lane = tid & 31;
  const int lo   = lane & 15;   // row-in-tile (A) / col-in-tile (B,C)
  const int hi   = lane >> 4;   // k-pair select (A,B) / row-half (C)
  const int m0   = (bm << 6) + (wave << 4);
  const int nb0  = bn << 6;

  v8fu acc[4] = {};

#pragma unroll 2
  for (int kk = 0; kk < K; kk += 4) {
    const int k0 = kk + (hi << 1);
    v2f a;
    if (TA == 0) {
      a = *(const v2f*)(A + (long)(m0 + lo) * lda + k0);      // b64 load
    } else {
      a.x = A[(long)k0 * lda + (m0 + lo)];
      a.y = A[(long)(k0 + 1) * lda + (m0 + lo)];
    }
#pragma unroll
    for (int j = 0; j < 4; ++j) {
      const int n0 = nb0 + (j << 4) + lo;
      v2f b;
      if (TB == 0) {
        b.x = B[(long)k0 * ldb + n0];
        b.y = B[(long)(k0 + 1) * ldb + n0];
      } else {
        b = *(const v2f*)(B + (long)n0 * ldb + k0);           // b64 load
      }
      acc[j].v = __builtin_amdgcn_wmma_f32_16x16x4_f32(
          false, a, false, b, (short)0, acc[j].v, false, false);
    }
  }

#pragma unroll
  for (int j = 0; j < 4; ++j) {
    const int n = nb0 + (j << 4) + lo;
#pragma unroll
    for (int v = 0; v < 8; ++v) {
      const int m = m0 + v + (hi << 3);
      float val = alpha * acc[j].f[v];
      if (E) val += beta * E[(long)m * lde + n];
      if (gam != 0.0f && m == n) val += gam;
      if (tc) C[(long)n * ldc + m] = val;
      else    C[(long)m * ldc + n] = val;
    }
  }
}

// ---------------------------------------------------------------------------
// In-LDS Gauss-Jordan inverse of an SPD matrix (stands in for cho_solve /
// pinv on SPD inputs). One matrix per workgroup (256 threads), n in {64,128}.
// Optional elementwise addend:  Out = inv(Ain + Add).  LDS: n*n*4 <= 64KB.
// ---------------------------------------------------------------------------
__global__ __launch_bounds__(256) void gj_inverse(
    const float* __restrict__ Ain, long sA,
    const float* __restrict__ Add, long sAdd,
    float* __restrict__ Out, long sOut, int n)
{
  __shared__ float Msh[NS * NS];
  const long b = blockIdx.x;
  const float* a  = Ain + b * sA;
  const float* ad = Add ? Add + b * sAdd : nullptr;
  float* o = Out + b * sOut;
  const int tid = threadIdx.x;
  const int nn  = n * n;

  for (int i = tid; i < nn; i += 256)
    Msh[i] = ad ? (a[i] + ad[i]) : a[i];
  __syncthreads();

  const int tpr   = 256 / n;       // threads per row: 2 (n=128) or 4 (n=64)
  const int row   = tid / tpr;
  const int sub   = tid - row * tpr;
  const int chunk = n / tpr;
  const int j0    = sub * chunk;

  for (int k = 0; k < n; ++k) {
    const float ip = 1.0f / Msh[k * n + k];  // LDS broadcast read
    const float f  = Msh[row * n + k];       // snapshot column k (pre-update)
    __syncthreads();
    if (row == k) {
      for (int j = j0; j < j0 + chunk; ++j) {
        const float v = (j == k) ? 1.0f : Msh[k * n + j];
        Msh[k * n + j] = v * ip;
      }
    }
    __syncthreads();
    if (row != k) {
      for (int j = j0; j < j0 + chunk; ++j) {
        const float v = (j == k) ? 0.0f : Msh[row * n + j];
        Msh[row * n + j] = v - f * Msh[k * n + j];
      }
    }
    __syncthreads();
  }

  for (int i = tid; i < nn; i += 256) o[i] = Msh[i];
}

// C := (C + C^T) / 2, batched over blockIdx.y
__global__ void sym_kernel(float* __restrict__ C, long sC, int n)
{
  float* M = C + (long)blockIdx.y * sC;
  const int idx = blockIdx.x * blockDim.x + threadIdx.x;
  const int i = idx / n, j = idx - i * n;
  if (i < n && j < n && i <= j) {
    const float v = 0.5f * (M[i * n + j] + M[j * n + i]);
    M[i * n + j] = v;
    M[j * n + i] = v;
  }
}

// Out = X + s*Y (elementwise)
__global__ void axpy_kernel(const float* __restrict__ X,
                            const float* __restrict__ Y,
                            float s, float* __restrict__ Out, int n)
{
  const int i = blockIdx.x * blockDim.x + threadIdx.x;
  if (i < n) Out[i] = X[i] + s * Y[i];
}

// out[i*cols + j] = vec[i]   (broadcast mu0 across the batch dimension)
__global__ void bcast_kernel(const float* __restrict__ vec,
                             float* __restrict__ out, int rows, int cols)
{
  const int idx = blockIdx.x * blockDim.x + threadIdx.x;
  if (idx < rows * cols) out[idx] = vec[idx / cols];
}

__global__ void zero_kernel(float* __restrict__ p, long n)
{
  const long i = (long)blockIdx.x * blockDim.x + threadIdx.x;
  if (i < n) p[i] = 0.0f;
}

// ---------------------------------------------------------------------------
// Host-side sequencing
// ---------------------------------------------------------------------------
static inline void gemm(hipStream_t st,
    const float* A, int lda, long sA, int ta,
    const float* B, int ldb, long sB, int tb,
    const float* E, int lde, long sE,
    float* C, int ldc, long sC, int tc,
    int M, int N, int K, float al, float be, float ga, int nb)
{
  dim3 g((unsigned)((M / 64) * (N / 64)), (unsigned)nb);
  if (!ta && !tb)
    gemm_wmma<0, 0><<<g, 128, 0, st>>>(A, lda, sA, B, ldb, sB, E, lde, sE,
                                       C, ldc, sC, tc, N, K, al, be, ga);
  else if (!ta && tb)
    gemm_wmma<0, 1><<<g, 128, 0, st>>>(A, lda, sA, B, ldb, sB, E, lde, sE,
                                       C, ldc, sC, tc, N, K, al, be, ga);
  else if (ta && !tb)
    gemm_wmma<1, 0><<<g, 128, 0, st>>>(A, lda, sA, B, ldb, sB, E, lde, sE,
                                       C, ldc, sC, tc, N, K, al, be, ga);
  else
    gemm_wmma<1, 1><<<g, 128, 0, st>>>(A, lda, sA, B, ldb, sB, E, lde, sE,
                                       C, ldc, sC, tc, N, K, al, be, ga);
}

static inline void gjinv(hipStream_t st, const float* Ain, long sA,
                         const float* Add, long sAdd,
                         float* Out, long sOut, int n, int nb)
{
  gj_inverse<<<nb, 256, 0, st>>>(Ain, sA, Add, sAdd, Out, sOut, n);
}

static inline void symm(hipStream_t st, float* C, long sC, int n, int nb)
{
  dim3 g((unsigned)((n * n + 255) / 256), (unsigned)nb);
  sym_kernel<<<g, 256, 0, st>>>(C, sC, n);
}

extern "C" void kernel_launch(void* const* d_in, const int* in_sizes, int n_in,
                              void* d_out, int out_size, void* d_ws, size_t ws_size,
                              hipStream_t stream)
{
  (void)in_sizes; (void)n_in; (void)out_size; (void)ws_size;
  const float* y    = (const float*)d_in[0];  // [B,T,O]
  const float* a    = (const float*)d_in[1];  // [N,N]
  const float* Lq   = (const float*)d_in[2];  // [N,N] lower-tri
  const float* L0   = (const float*)d_in[3];  // [N,N] lower-tri
  const float* mu0  = (const float*)d_in[4];  // [N]
  const float* Bm   = (const float*)d_in[5];  // [O,N]
  const float* Lr   = (const float*)d_in[6];  // [O,O] lower-tri

  float* out = (float*)d_out;
  float* out_mut  = out;                                  // [B,T,N]
  float* out_ptT  = out_mut + (long)B_SZ * T_LEN * NS;    // [T,N,N]
  float* out_yhat = out_ptT + (long)T_LEN * NS * NS;      // [B,T,O]
  float* out_sig  = out_yhat + (long)B_SZ * T_LEN * NO;   // [T,O,O]
  float* out_hs   = out_sig + (long)T_LEN * NO * NO;      // [T-1,N,N]

  // workspace allocator
  float* ws = (float*)d_ws;
  size_t off = 0;
  auto alloc = [&](size_t ne) { float* p = ws + off; off += ne; return p; };

  const long NN = (long)NS * NS;       // 16384
  const long NO2 = (long)NO * NO;      // 4096
  const long NB = (long)NS * B_SZ;     // 32768

  float* q      = alloc(NN);
  float* r      = alloc(NO2);
  float* p0     = alloc(NN);
  float* qinv   = alloc(NN);
  float* rinv   = alloc(NO2);
  float* bTrinv = alloc((long)NS * NO);
  float* bTrB   = alloc(NN);
  float* qbsum  = alloc(NN);           // qinv + bTrB
  float* siginv = alloc(NO2);
  float* pbT    = alloc((long)NS * NO);
  float* tmp1   = alloc(NN);
  float* piinv  = alloc(NN);
  float* piq    = alloc(NN);
  float* Stmp   = alloc(NN);
  float* t2     = alloc(NN);
  float* ktmp   = alloc(NB);
  float* ptt    = alloc((long)T_LEN * NN);   // p_tt_all
  float* ppred  = alloc((long)T_LEN * NN);   // p_pred_all (t>=1 valid)
  float* imgb   = alloc((long)T_LEN * NN);
  float* gall   = alloc((long)T_LEN * NS * NO);
  float* piqA   = alloc((long)T_LEN * NN);   // (piinv*qinv)*A per y-step
  float* gam    = alloc((long)T_LEN * NN);   // gamma_all
  float* mutt   = alloc((long)T_LEN * NB);   // mu_tt  as [T][N][B]
  float* mupr   = alloc((long)T_LEN * NB);   // mu_ttm1 as [T][N][B]
  float* ksi    = alloc((long)T_LEN * NB);   // ksi_all as [T][N][B]
  float* pinvA  = alloc((long)T_LEN * NN);   // p_inv_all
  float* ppinv  = alloc((long)(T_LEN - 1) * NN);
  float* hstmp  = alloc((long)(T_LEN - 1) * NN);
  float* t3     = alloc((long)T_LEN * NB);

  const int ldY = T_LEN * NO;   // batch stride inside y when sliced at time t

  // ---- Phase 0: static precompute ----
  gemm(stream, Lq, NS, 0, 0, Lq, NS, 0, 1, nullptr, 0, 0, q,  NS, 0, 0, NS, NS, NS, 1, 0, 0, 1);
  gemm(stream, Lr, NO, 0, 0, Lr, NO, 0, 1, nullptr, 0, 0, r,  NO, 0, 0, NO, NO, NO, 1, 0, 0, 1);
  gemm(stream, L0, NS, 0, 0, L0, NS, 0, 1, nullptr, 0, 0, p0, NS, 0, 0, NS, NS, NS, 1, 0, 0, 1);
  gjinv(stream, q, 0, nullptr, 0, qinv, 0, NS, 1);
  gjinv(stream, r, 0, nullptr, 0, rinv, 0, NO, 1);
  gemm(stream, Bm, NS, 0, 1, rinv, NO, 0, 0, nullptr, 0, 0, bTrinv, NO, 0, 0, NS, NO, NO, 1, 0, 0, 1);
  gemm(stream, bTrinv, NO, 0, 0, Bm, NS, 0, 0, nullptr, 0, 0, bTrB, NS, 0, 0, NS, NS, NO, 1, 0, 0, 1);
  axpy_kernel<<<(NN + 255) / 256, 256, 0, stream>>>(qinv, bTrB, 1.0f, qbsum, (int)NN);

  // ---- Phase 1: forward covariance chain (sequential over t) ----
  for (int t = 0; t < T_LEN; ++t) {
    const float* pp;  // p_{t|t-1}
    if (t == 0) {
      pp = p0;
    } else {
      // tmp1 = p_tt[t-1] * A^T ; p_pred[t] = q + A*tmp1
      gemm(stream, ptt + (long)(t - 1) * NN, NS, 0, 0, a, NS, 0, 1, nullptr, 0, 0,
           tmp1, NS, 0, 0, NS, NS, NS, 1, 0, 0, 1);
      gemm(stream, a, NS, 0, 0, tmp1, NS, 0, 0, q, NS, 0,
           ppred + (long)t * NN, NS, 0, 0, NS, NS, NS, 1, 1, 0, 1);
      pp = ppred + (long)t * NN;
    }
    // pbT = pp * B^T ; sig = r + B*pbT (symmetrized)
    gemm(stream, pp, NS, 0, 0, Bm, NS, 0, 1, nullptr, 0, 0, pbT, NO, 0, 0, NS, NO, NS, 1, 0, 0, 1);
    gemm(stream, Bm, NS, 0, 0, pbT, NO, 0, 0, r, NO, 0,
         out_sig + (long)t * NO2, NO, 0, 0, NO, NO, NS, 1, 1, 0, 1);
    symm(stream, out_sig + (long)t * NO2, 0, NO, 1);
    // g = pbT * sig^{-1}
    gjinv(stream, out_sig + (long)t * NO2, 0, nullptr, 0, siginv, 0, NO, 1);
    gemm(stream, pbT, NO, 0, 0, siginv, NO, 0, 0, nullptr, 0, 0,
         gall + (long)t * NS * NO, NO, 0, 0, NS, NO, NO, 1, 0, 0, 1);
    // imgb = I - g*B ; p_tt = sym(imgb * pp)
    gemm(stream, gall + (long)t * NS * NO, NO, 0, 0, Bm, NS, 0, 0, nullptr, 0, 0,
         imgb + (long)t * NN, NS, 0, 0, NS, NS, NO, -1, 0, 1, 1);
    gemm(stream, imgb + (long)t * NN, NS, 0, 0, pp, NS, 0, 0, nullptr, 0, 0,
         ptt + (long)t * NN, NS, 0, 0, NS, NS, NS, 1, 0, 0, 1);
    symm(stream, ptt + (long)t * NN, 0, NS, 1);
  }

  // ---- Phase 2: hs[i] = p_tt[i] * A^T * inv(p_pred[i+1])  (batch-parallel) ----
  gjinv(stream, ppred + NN, NN, nullptr, 0, ppinv, NN, NS, T_LEN - 1);
  gemm(stream, ptt, NS, NN, 0, a, NS, 0, 1, nullptr, 0, 0,
       hstmp, NS, NN, 0, NS, NS, NS, 1, 0, 0, T_LEN - 1);
  gemm(stream, hstmp, NS, NN, 0, ppinv, NS, NN, 0, nullptr, 0, 0,
       out_hs, NS, NN, 0, NS, NS, NS, 1, 0, 0, T_LEN - 1);

  // ---- Phase 3: batched mean recursion (sequential over t, batch via N=256) ----
  bcast_kernel<<<(int)((NB + 255) / 256), 256, 0, stream>>>(mu0, mupr, NS, B_SZ);
  for (int t = 0; t < T_LEN; ++t) {
    if (t > 0) {  // mu_pred[t] = A * mu_tt[t-1]
      gemm(stream, a, NS, 0, 0, mutt + (long)(t - 1) * NB, B_SZ, 0, 0, nullptr, 0, 0,
           mupr + (long)t * NB, B_SZ, 0, 0, NS, B_SZ, NS, 1, 0, 0, 1);
    }
    // mu_tt[t] = imgb[t]*mu_pred[t] ;  += g[t]*Y_t
    gemm(stream, imgb + (long)t * NN, NS, 0, 0, mupr + (long)t * NB, B_SZ, 0, 0,
         nullptr, 0, 0, mutt + (long)t * NB, B_SZ, 0, 0, NS, B_SZ, NS, 1, 0, 0, 1);
    gemm(stream, gall + (long)t * NS * NO, NO, 0, 0, y + (long)t * NO, ldY, 0, 1,
         mutt + (long)t * NB, B_SZ, 0, mutt + (long)t * NB, B_SZ, 0, 0,
         NS, B_SZ, NO, 1, 1, 0, 1);
  }
  // yhat[b,t,:] = B * mu_pred[t]  (batched over t, transposed store)
  gemm(stream, Bm, NS, 0, 0, mupr, B_SZ, NB, 0, nullptr, 0, 0,
       out_yhat, ldY, NO, 1, NO, B_SZ, NS, 1, 0, 0, T_LEN);

  // ---- Phase 4: backward gamma chain (sequential) ----
  zero_kernel<<<(int)((NN + 255) / 256), 256, 0, stream>>>(gam + (long)(T_LEN - 1) * NN, NN);
  for (int ty = T_LEN - 1; ty >= 1; --ty) {
    // pi = gamma[ty] + bTrB + qinv ; piinv = inv(pi)
    gjinv(stream, gam + (long)ty * NN, 0, qbsum, 0, piinv, 0, NS, 1);
    gemm(stream, piinv, NS, 0, 0, qinv, NS, 0, 0, nullptr, 0, 0, piq, NS, 0, 0, NS, NS, NS, 1, 0, 0, 1);
    gemm(stream, piq, NS, 0, 0, a, NS, 0, 0, nullptr, 0, 0,
         piqA + (long)ty * NN, NS, 0, 0, NS, NS, NS, 1, 0, 0, 1);
    // S = A - piq*A ; gamma[ty-1] = A^T * (qinv * S)
    axpy_kernel<<<(int)((NN + 255) / 256), 256, 0, stream>>>(a, piqA + (long)ty * NN, -1.0f, Stmp, (int)NN);
    gemm(stream, qinv, NS, 0, 0, Stmp, NS, 0, 0, nullptr, 0, 0, t2, NS, 0, 0, NS, NS, NS, 1, 0, 0, 1);
    gemm(stream, a, NS, 0, 1, t2, NS, 0, 0, nullptr, 0, 0,
         gam + (long)(ty - 1) * NN, NS, 0, 0, NS, NS, NS, 1, 0, 0, 1);
  }

  // ---- Phase 5: backward ksi chain (sequential over t, batched over B) ----
  zero_kernel<<<(int)((NB + 255) / 256), 256, 0, stream>>>(ksi + (long)(T_LEN - 1) * NB, NB);
  for (int ty = T_LEN - 1; ty >= 1; --ty) {
    // ktmp = ksi[ty] + bTrinv * Y_ty ; ksi[ty-1] = piqA[ty]^T * ktmp
    gemm(stream, bTrinv, NO, 0, 0, y + (long)ty * NO, ldY, 0, 1,
         ksi + (long)ty * NB, B_SZ, 0, ktmp, B_SZ, 0, 0, NS, B_SZ, NO, 1, 1, 0, 1);
    gemm(stream, piqA + (long)ty * NN, NS, 0, 1, ktmp, B_SZ, 0, 0, nullptr, 0, 0,
         ksi + (long)(ty - 1) * NB, B_SZ, 0, 0, NS, B_SZ, NS, 1, 0, 0, 1);
  }

  // ---- Phase 6: merge (batch-parallel over t) ----
  gjinv(stream, ptt, NN, nullptr, 0, pinvA, NN, NS, T_LEN);            // p_inv
  gjinv(stream, gam, NN, pinvA, NN, out_ptT, NN, NS, T_LEN);           // ptT
  gemm(stream, pinvA, NS, NN, 0, mutt, B_SZ, NB, 0, ksi, B_SZ, NB,
       t3, B_SZ, NB, 0, NS, B_SZ, NS, 1, 1, 0, T_LEN);                 // t3 = ksi + p_inv*mu_tt
  gemm(stream, out_ptT, NS, NN, 0, t3, B_SZ, NB, 0, nullptr, 0, 0,
       out_mut + 0, T_LEN * NS, NS, 1, NS, B_SZ, NS, 1, 0, 0, T_LEN);  // mutT -> [B,T,N]
}